// MySAGEConv_72103910966001
// MI455X (gfx1250) — compile-verified
//
#include <hip/hip_runtime.h>

#define P_     16
#define K_     256
#define W_     8
#define F_IN   128
#define F_OUT  128
#define TILE_M 16

typedef __attribute__((ext_vector_type(2))) float v2f;
typedef __attribute__((ext_vector_type(8))) float v8f;

__global__ __launch_bounds__(256) void sage_fused_kernel(
    const int*   __restrict__ codes,     // (N_SRC, 16)
    const int*   __restrict__ indices,   // (E,)
    const int*   __restrict__ indptr,    // (N_DST+1,)
    const float* __restrict__ h_self,    // (N_DST, 128)
    const float* __restrict__ codebook,  // (16, 256, 8)
    const float* __restrict__ W_neigh,   // (128, 128)
    const float* __restrict__ W_self,    // (128, 128)
    const float* __restrict__ b_self,    // (128,)
    float*       __restrict__ out,       // (N_DST, 128)
    int n_dst)
{
    __shared__ __align__(16) float lds_hn[TILE_M * F_IN]; // mean-aggregated neighbor feats
    __shared__ __align__(16) float lds_hs[TILE_M * F_IN]; // self feats tile
    __shared__ int   sidx[TILE_M * 16];
    __shared__ float sinv[TILE_M];

    const int tid      = threadIdx.x;
    const int row_base = blockIdx.x * TILE_M;

    // ---- Stage edge lists: 16 rows x up-to-16 neighbor source ids ----
    {
        const int row = tid >> 4;
        const int e   = tid & 15;
        int deg = 0, src = -1;
        if (row_base + row < n_dst) {
            const int beg = indptr[row_base + row];
            const int end = indptr[row_base + row + 1];
            deg = end - beg;
            if (e < deg) src = indices[beg + e];
        }
        sidx[tid] = src;
        if (e == 0) sinv[row] = 1.0f / fmaxf((float)deg, 1.0f);
    }
    __syncthreads();

    // ---- Phase A: decode + mean-aggregate h_neigh, and load h_self tile ----
    // Each item = one (row, 4-wide column group). 16*32 = 512 items, 2 per thread.
    for (int idx = tid; idx < TILE_M * (F_IN / 4); idx += 256) {
        const int row = idx >> 5;      // 0..15
        const int c4  = idx & 31;      // float4 group in the 128-wide row
        const int gr  = row_base + row;

        float4 hs = make_float4(0.f, 0.f, 0.f, 0.f);
        if (gr < n_dst)
            hs = *(const float4*)(h_self + (size_t)gr * F_IN + c4 * 4);
        *(float4*)(&lds_hs[row * F_IN + c4 * 4]) = hs;

        // columns c4*4 .. c4*4+3 all live in sub-codebook p, word offset w0
        const int p  = c4 >> 1;
        const int w0 = (c4 & 1) * 4;
        float ax = 0.f, ay = 0.f, az = 0.f, aw = 0.f;
        #pragma unroll
        for (int e = 0; e < 16; ++e) {
            const int s = sidx[row * 16 + e];
            if (s >= 0) {
                const int code = codes[s * P_ + p];
                const float4 cb =
                    *(const float4*)(codebook + ((p * K_ + code) * W_ + w0));
                ax += cb.x; ay += cb.y; az += cb.z; aw += cb.w;
            }
        }
        const float inv = sinv[row];
        const float4 hn = make_float4(ax * inv, ay * inv, az * inv, aw * inv);
        *(float4*)(&lds_hn[row * F_IN + c4 * 4]) = hn;
    }
    __syncthreads();

    // ---- Phase B: out(16x128) = hn @ Wn^T + hs @ Ws^T + b via V_WMMA_F32_16X16X4_F32 ----
    const int lane  = tid & 31;
    const int wave  = tid >> 5;          // 0..7 -> 16-wide N tile
    const int nbase = wave * 16;
    const int m     = lane & 15;         // M row (A) / N col (B,C,D)
    const int khalf = (lane >> 4) * 2;   // lanes 0-15: K=0,1 ; lanes 16-31: K=2,3

    v8f acc = {};
    const float* an = lds_hn + m * F_IN + khalf;
    const float* as = lds_hs + m * F_IN + khalf;
    const float* wn = W_neigh + (size_t)(nbase + m) * F_IN + khalf;
    const float* ws = W_self  + (size_t)(nbase + m) * F_IN + khalf;

    #pragma unroll 8
    for (int k0 = 0; k0 < F_IN; k0 += 4) {
        v2f a = *(const v2f*)(an + k0);
        v2f b = *(const v2f*)(wn + k0);
        acc = __builtin_amdgcn_wmma_f32_16x16x4_f32(
            false, a, false, b, (short)0, acc, false, false);
    }
    #pragma unroll 8
    for (int k0 = 0; k0 < F_IN; k0 += 4) {
        v2f a = *(const v2f*)(as + k0);
        v2f b = *(const v2f*)(ws + k0);
        acc = __builtin_amdgcn_wmma_f32_16x16x4_f32(
            false, a, false, b, (short)0, acc, false, false);
    }

    // Epilogue: C/D layout -> VGPR j holds row j (lanes 0-15) / row j+8 (lanes 16-31)
    const float bias  = b_self[nbase + m];
    const int   rhalf = (lane >> 4) * 8;
    #pragma unroll
    for (int j = 0; j < 8; ++j) {
        const int gr = row_base + rhalf + j;
        if (gr < n_dst)
            out[(size_t)gr * F_OUT + nbase + m] = acc[j] + bias;
    }
}

extern "C" void kernel_launch(void* const* d_in, const int* in_sizes, int n_in,
                              void* d_out, int out_size, void* d_ws, size_t ws_size,
                              hipStream_t stream) {
    (void)n_in; (void)out_size; (void)d_ws; (void)ws_size;
    const int*   codes    = (const int*)  d_in[0];
    const int*   indices  = (const int*)  d_in[1];
    const int*   indptr   = (const int*)  d_in[2];
    const float* h_self   = (const float*)d_in[3];
    const float* codebook = (const float*)d_in[4];
    const float* W_neigh  = (const float*)d_in[5];
    const float* W_self   = (const float*)d_in[6];
    const float* b_self   = (const float*)d_in[7];
    float*       out      = (float*)d_out;

    const int n_dst  = in_sizes[2] - 1;              // indptr has N_DST+1 entries
    const int blocks = (n_dst + TILE_M - 1) / TILE_M;

    sage_fused_kernel<<<blocks, 256, 0, stream>>>(
        codes, indices, indptr, h_self, codebook, W_neigh, W_self, b_self,
        out, n_dst);
}